// DiagnosticPhysics_33320356282807
// MI455X (gfx1250) — compile-verified
//
#include <hip/hip_runtime.h>
#include <stdint.h>

#define RDRY 287.0f
#define EPSC 1e-8f

constexpr int NZ37 = 37;
constexpr int TILE = 128;   // threads per block == columns per block (4 wave32s)

// ---- gfx1250 async global<->LDS copy helpers ------------------------------
#if defined(__gfx1250__) && __has_builtin(__builtin_amdgcn_global_load_async_to_lds_b128)
#define USE_ASYNC_LD128 1
#endif
#if defined(__gfx1250__) && __has_builtin(__builtin_amdgcn_global_load_async_to_lds_b32)
#define USE_ASYNC_LD32 1
#endif
#if defined(__gfx1250__) && __has_builtin(__builtin_amdgcn_global_store_async_from_lds_b128)
#define USE_ASYNC_ST128 1
#endif
#if defined(__gfx1250__) && __has_builtin(__builtin_amdgcn_global_store_async_from_lds_b32)
#define USE_ASYNC_ST32 1
#endif

// Types matching the builtin signatures (confirmed by clang diagnostics):
//   b128: 'int __attribute__((vector_size(16))) __device__ *' (addrspace(1))
//   b32 : 'int __device__ *'
typedef int v4i __attribute__((vector_size(16)));
typedef __attribute__((address_space(1))) v4i gv4i;   // global int4
typedef __attribute__((address_space(3))) v4i lv4i;   // LDS int4
typedef __attribute__((address_space(1))) int gi32;   // global int
typedef __attribute__((address_space(3))) int li32;   // LDS int

__device__ __forceinline__ void ld16_g2l(const float* g, float* l) {
#if defined(USE_ASYNC_LD128)
  __builtin_amdgcn_global_load_async_to_lds_b128(
      (gv4i*)(v4i*)g, (lv4i*)(v4i*)l, 0, 0);
#else
  *(float4*)l = *(const float4*)g;
#endif
}
__device__ __forceinline__ void ld4_g2l(const float* g, float* l) {
#if defined(USE_ASYNC_LD32)
  __builtin_amdgcn_global_load_async_to_lds_b32(
      (gi32*)(int*)g, (li32*)(int*)l, 0, 0);
#else
  *l = *g;
#endif
}
__device__ __forceinline__ void st16_l2g(float* g, const float* l) {
#if defined(USE_ASYNC_ST128)
  __builtin_amdgcn_global_store_async_from_lds_b128(
      (gv4i*)(v4i*)g, (lv4i*)(v4i*)l, 0, 0);
#else
  *(float4*)g = *(const float4*)l;
#endif
}
__device__ __forceinline__ void st4_l2g(float* g, const float* l) {
#if defined(USE_ASYNC_ST32)
  __builtin_amdgcn_global_store_async_from_lds_b32(
      (gi32*)(int*)g, (li32*)(int*)l, 0, 0);
#else
  *g = *l;
#endif
}

__device__ __forceinline__ void async_wait0() {
#if defined(USE_ASYNC_LD128) || defined(USE_ASYNC_LD32) || \
    defined(USE_ASYNC_ST128) || defined(USE_ASYNC_ST32)
#if __has_builtin(__builtin_amdgcn_s_wait_asynccnt)
  __builtin_amdgcn_s_wait_asynccnt(0);
#else
  asm volatile("s_wait_asynccnt 0" ::: "memory");
#endif
#endif
}

// Coalesced LDS tile -> linear global chunk (async store path).
__device__ __forceinline__ void store_tile_l2g(float* gdst, const float* lsrc,
                                               int nflt, int tid) {
  if ((((uintptr_t)gdst) & 15u) == 0) {
    const int n4 = nflt >> 2;
    for (int i = tid; i < n4; i += TILE) st16_l2g(gdst + 4 * i, lsrc + 4 * i);
    for (int i = (n4 << 2) + tid; i < nflt; i += TILE) st4_l2g(gdst + i, lsrc + i);
  } else {
    for (int i = tid; i < nflt; i += TILE) st4_l2g(gdst + i, lsrc + i);
  }
}

// ---- specialized kernel: NZ == 37 -----------------------------------------
__global__ __launch_bounds__(TILE) void diag37_kernel(
    const float* __restrict__ ps, const float* __restrict__ T,
    const float* __restrict__ dv, const float* __restrict__ ak,
    const float* __restrict__ bk, const float* __restrict__ dsg,
    float* __restrict__ out, int C) {
  // sT: temperature in, pressure out (in place)
  // sD: divergence in, omega out (in place)
  // sH: geopotential out
  __shared__ __align__(16) float sT[TILE * NZ37];
  __shared__ __align__(16) float sD[TILE * NZ37];
  __shared__ __align__(16) float sH[TILE * NZ37];

  const int tid  = threadIdx.x;
  const int col0 = blockIdx.x * TILE;
  const int tcols = (C - col0 < TILE) ? (C - col0) : TILE;
  const int nflt  = tcols * NZ37;

  // ---- stage inputs: linear coalesced async global->LDS -------------------
  const float* gT = T  + (size_t)col0 * NZ37;   // 16B aligned: col0%128==0
  const float* gD = dv + (size_t)col0 * NZ37;
  const int n4 = nflt >> 2;
  for (int i = tid; i < n4; i += TILE) {
    ld16_g2l(gT + 4 * i, &sT[4 * i]);
    ld16_g2l(gD + 4 * i, &sD[4 * i]);
  }
  for (int i = (n4 << 2) + tid; i < nflt; i += TILE) {
    ld4_g2l(gT + i, &sT[i]);
    ld4_g2l(gD + i, &sD[i]);
  }
  async_wait0();
  __syncthreads();

  // ---- per-column compute (own column only: read-then-write, no hazards) --
  const int col = col0 + tid;
  if (col < C) {
    const float psv   = ps[col];
    const float invps = 1.0f / (psv + EPSC);
    // stride 37 (odd) across lanes -> conflict-free on 64 LDS banks
    float* tc = &sT[tid * NZ37];   // reads T, writes p
    float* dc = &sD[tid * NZ37];   // reads div, writes omega
    float* hc = &sH[tid * NZ37];   // writes phi

    // omega: column integral, then inclusive cumsum (in-place into sD)
    float colint = 0.0f;
#pragma unroll
    for (int k = 0; k < NZ37; ++k) colint += psv * dc[k] * dsg[k];
    float run = 0.0f;
#pragma unroll
    for (int k = 0; k < NZ37; ++k) {
      run += psv * dc[k] * dsg[k];
      dc[k] = (bk[k] * colint - run) * invps;
    }

    // pressure + geopotential: downward streaming scan (p in-place into sT)
    float p_hi = ak[NZ37 - 1] + bk[NZ37 - 1] * psv;
    float T_hi = tc[NZ37 - 1];
    tc[NZ37 - 1] = p_hi;
    hc[NZ37 - 1] = 0.0f;
    float acc = 0.0f;
#pragma unroll
    for (int k = NZ37 - 2; k >= 0; --k) {
      const float p_lo = ak[k] + bk[k] * psv;
      const float T_lo = tc[k];
      const float r = (p_hi + EPSC) / (p_lo + EPSC);
      acc += RDRY * (0.5f * (T_hi + T_lo)) * logf(r);
      tc[k] = p_lo;
      hc[k] = acc;
      p_hi = p_lo;
      T_hi = T_lo;
    }
  }
  __syncthreads();

  // ---- drain results: linear coalesced async LDS->global ------------------
  const size_t cz = (size_t)C * NZ37;
  float* gp = out + (size_t)col0 * NZ37;            // pressure
  float* gh = out + cz + (size_t)col0 * NZ37;       // geopotential
  float* gw = out + 2 * cz + (size_t)col0 * NZ37;   // omega
  store_tile_l2g(gp, sT, nflt, tid);
  store_tile_l2g(gh, sH, nflt, tid);
  store_tile_l2g(gw, sD, nflt, tid);
  // s_endpgm performs an implicit wait-idle, draining ASYNCcnt.
}

// ---- generic fallback (any NZ), direct global streaming -------------------
__global__ void diag_generic_kernel(
    const float* __restrict__ ps, const float* __restrict__ T,
    const float* __restrict__ dv, const float* __restrict__ ak,
    const float* __restrict__ bk, const float* __restrict__ dsg,
    float* __restrict__ out, int C, int nz) {
  const int col = blockIdx.x * blockDim.x + threadIdx.x;
  if (col >= C) return;
  const float psv   = ps[col];
  const float invps = 1.0f / (psv + EPSC);
  const float* tc = T  + (size_t)col * nz;
  const float* dc = dv + (size_t)col * nz;
  const size_t obase = (size_t)col * nz;
  float* op = out + obase;
  float* oh = out + (size_t)C * nz + obase;
  float* ow = out + 2 * (size_t)C * nz + obase;

  float colint = 0.0f;
  for (int k = 0; k < nz; ++k) colint += psv * dc[k] * dsg[k];
  float run = 0.0f;
  for (int k = 0; k < nz; ++k) {
    run += psv * dc[k] * dsg[k];
    ow[k] = (bk[k] * colint - run) * invps;
  }
  float p_hi = ak[nz - 1] + bk[nz - 1] * psv;
  float T_hi = tc[nz - 1];
  op[nz - 1] = p_hi;
  oh[nz - 1] = 0.0f;
  float acc = 0.0f;
  for (int k = nz - 2; k >= 0; --k) {
    const float p_lo = ak[k] + bk[k] * psv;
    const float T_lo = tc[k];
    acc += RDRY * (0.5f * (T_hi + T_lo)) * logf((p_hi + EPSC) / (p_lo + EPSC));
    op[k] = p_lo;
    oh[k] = acc;
    p_hi = p_lo;
    T_hi = T_lo;
  }
}

extern "C" void kernel_launch(void* const* d_in, const int* in_sizes, int n_in,
                              void* d_out, int out_size, void* d_ws, size_t ws_size,
                              hipStream_t stream) {
  const float* ps  = (const float*)d_in[0];
  const float* T   = (const float*)d_in[1];
  const float* dv  = (const float*)d_in[2];
  const float* ak  = (const float*)d_in[3];
  const float* bk  = (const float*)d_in[4];
  const float* dsg = (const float*)d_in[5];
  float* out = (float*)d_out;

  const int C  = in_sizes[0];   // B*NY*NX columns
  const int nz = in_sizes[3];   // NZ from a_k

  if (nz == NZ37) {
    const int grid = (C + TILE - 1) / TILE;
    diag37_kernel<<<grid, TILE, 0, stream>>>(ps, T, dv, ak, bk, dsg, out, C);
  } else {
    const int grid = (C + 255) / 256;
    diag_generic_kernel<<<grid, 256, 0, stream>>>(ps, T, dv, ak, bk, dsg, out, C, nz);
  }
}